// UniGATConv_82128364634688
// MI455X (gfx1250) — compile-verified
//
#include <hip/hip_runtime.h>
#include <math.h>

// Problem constants (match reference)
#define NV 50000
#define EI 600000
#define ME 100000
#define INC 256
#define HH 8
#define CC 32
#define HC 256   // H*C

typedef float v2f __attribute__((ext_vector_type(2)));
typedef float v8f __attribute__((ext_vector_type(8)));

// ---- workspace layout (in floats) ----
#define OFF_X0   ((size_t)0)                         // [NV, HC]
#define OFF_XE   (OFF_X0 + (size_t)NV * HC)          // [ME, HC]
#define OFF_XV   (OFF_XE + (size_t)ME * HC)          // [NV, HC]
#define OFF_CNT  (OFF_XV + (size_t)NV * HC)          // [ME]
#define OFF_ASUM (OFF_CNT + (size_t)ME)              // [NV, HH]
#define OFF_AMAX (OFF_ASUM + (size_t)NV * HH)        // [NV, HH]
#define OFF_AE   (OFF_AMAX + (size_t)NV * HH)        // [ME, HH]   alpha per hyperedge
#define OFF_AEXP (OFF_AE + (size_t)ME * HH)          // [EI, HH]   alpha (then exp) per incidence
// zero region: Xe, Xv, cnt, asum (contiguous)
#define ZERO_N   ((size_t)ME * HC + (size_t)NV * HC + (size_t)ME + (size_t)NV * HH)

// LDS panel: 32 K-rows x 256 cols, padded row stride (272 % 64 == 16 -> bank-conflict-free
// B-fragment reads: lanes 0-15 hit banks b..b+15, lanes 16-31 hit banks b+32..b+47)
#define LDSROW 272

// ------------------- init -------------------
__global__ void k_init(float* __restrict__ ws) {
  size_t i = (size_t)blockIdx.x * blockDim.x + threadIdx.x;
  size_t stride = (size_t)gridDim.x * blockDim.x;
  for (size_t j = i; j < ZERO_N; j += stride) ws[OFF_XE + j] = 0.0f;
  for (size_t j = i; j < (size_t)NV * HH; j += stride) ws[OFF_AMAX + j] = -INFINITY;
}

// ------------- fp32 WMMA GEMM: X0 = X @ W, W staged via async global->LDS -------------
// grid.x = NV/16, block = 256 (8 waves); wave w covers output cols [32w, 32w+32)
__global__ void __launch_bounds__(256) k_gemm(const float* __restrict__ X,
                                              const float* __restrict__ W,
                                              float* __restrict__ X0) {
  __shared__ float Wl[32 * LDSROW];   // 34,816 bytes

  const int m0    = blockIdx.x << 4;
  const int wave  = threadIdx.x >> 5;
  const int lane  = threadIdx.x & 31;
  const int t     = threadIdx.x;
  const int n0    = wave << 5;
  const int mrow  = m0 + (lane & 15);
  const int khalf = (lane >> 4) << 1;     // lanes 0-15: K=0,1  lanes 16-31: K=2,3
  const int ncol  = n0 + (lane & 15);

  v8f acc0 = {0.f,0.f,0.f,0.f,0.f,0.f,0.f,0.f};
  v8f acc1 = {0.f,0.f,0.f,0.f,0.f,0.f,0.f,0.f};
  const float* arow = X + (size_t)mrow * INC;

  for (int kc = 0; kc < INC; kc += 32) {
    // ---- async copy of W[kc..kc+32) x 256 into padded LDS (b128 per lane) ----
#pragma unroll
    for (int j = 0; j < 8; ++j) {
      const int f   = j * 256 + t;   // 16-byte unit index within the 32KB panel
      const int row = f >> 6;        // 64 x 16B per K-row
      const int c16 = f & 63;
      const float* gsrc = W + ((size_t)(kc + row) << 8) + (c16 << 2);
      const unsigned ldst = (unsigned)(size_t)&Wl[row * LDSROW + (c16 << 2)];
      asm volatile("global_load_async_to_lds_b128 %0, %1, off"
                   :: "v"(ldst), "v"(gsrc) : "memory");
    }
    asm volatile("s_wait_asynccnt 0x0" ::: "memory");
    __syncthreads();

    // ---- 8 k-steps of fp32 WMMA with B fragments from LDS ----
#pragma unroll
    for (int kk = 0; kk < 32; kk += 4) {
      v2f a = *(const v2f*)(arow + kc + kk + khalf);
      const float* wl = &Wl[(kk + khalf) * LDSROW];
      v2f b0, b1;
      b0.x = wl[ncol];           b0.y = wl[LDSROW + ncol];
      b1.x = wl[ncol + 16];      b1.y = wl[LDSROW + ncol + 16];
      acc0 = __builtin_amdgcn_wmma_f32_16x16x4_f32(false, a, false, b0, (short)0, acc0, false, false);
      acc1 = __builtin_amdgcn_wmma_f32_16x16x4_f32(false, a, false, b1, (short)0, acc1, false, false);
    }
    __syncthreads();
  }

  // C/D layout: VGPR v: lanes 0-15 -> M=v, lanes 16-31 -> M=v+8; N = lane%16
  const int mc = m0 + ((lane >> 4) << 3);
  const int nc = n0 + (lane & 15);
#pragma unroll
  for (int v = 0; v < 8; ++v) {
    X0[(size_t)(mc + v) * HC + nc]      = acc0[v];
    X0[(size_t)(mc + v) * HC + nc + 16] = acc1[v];
  }
}

// ------------- scatter incidences -> hyperedges -------------
__global__ void __launch_bounds__(256) k_scatter_edge(const float* __restrict__ X0,
    const int* __restrict__ vertex, const int* __restrict__ edges,
    float* __restrict__ Xe, float* __restrict__ cnt) {
  const int e = blockIdx.x;
  const int t = threadIdx.x;
  const int v = vertex[e];
  const int m = edges[e];
  atomicAdd(&Xe[(size_t)m * HC + t], X0[(size_t)v * HC + t]);
  if (t == 0) atomicAdd(&cnt[m], 1.0f);
}

// ------------- normalize Xe by count; alpha_e = <Xe, att_e> per head -------------
__global__ void __launch_bounds__(256) k_edge_alpha(float* __restrict__ Xe,
    const float* __restrict__ cnt, const float* __restrict__ att,
    float* __restrict__ alpha_e) {
  const int m = blockIdx.x;
  const int t = threadIdx.x;
  const int h = t >> 5;
  const int lane = t & 31;
  const float inv = 1.0f / fmaxf(cnt[m], 1.0f);
  float x = Xe[(size_t)m * HC + t] * inv;
  Xe[(size_t)m * HC + t] = x;
  float p = x * att[t];
#pragma unroll
  for (int off = 16; off > 0; off >>= 1) p += __shfl_xor(p, off, 32);
  if (lane == 0) alpha_e[(size_t)m * HH + h] = p;
}

// ------------- leaky relu + segment max over vertices -------------
__global__ void k_leaky_amax(const float* __restrict__ alpha_e,
    const int* __restrict__ vertex, const int* __restrict__ edges,
    float* __restrict__ alphaE, float* __restrict__ amax) {
  size_t i = (size_t)blockIdx.x * blockDim.x + threadIdx.x;
  if (i >= (size_t)EI * HH) return;
  const int e = (int)(i >> 3);
  const int h = (int)(i & 7);
  float a = alpha_e[(size_t)edges[e] * HH + h];
  a = (a >= 0.0f) ? a : 0.2f * a;
  alphaE[i] = a;
  float* addr = &amax[(size_t)vertex[e] * HH + h];
  // monotonic float-max via int/uint atomics (amax initialized to -inf)
  if (a >= 0.0f) atomicMax((int*)addr, __float_as_int(a));
  else           atomicMin((unsigned int*)addr, __float_as_uint(a));
}

// ------------- exp(alpha - amax) + segment sum -------------
__global__ void k_exp_asum(float* __restrict__ alphaE,
    const int* __restrict__ vertex, const float* __restrict__ amax,
    float* __restrict__ asum) {
  size_t i = (size_t)blockIdx.x * blockDim.x + threadIdx.x;
  if (i >= (size_t)EI * HH) return;
  const int e = (int)(i >> 3);
  const int h = (int)(i & 7);
  const size_t vi = (size_t)vertex[e] * HH + h;
  float ex = expf(alphaE[i] - amax[vi]);
  alphaE[i] = ex;   // overwrite with exp for reuse
  atomicAdd(&asum[vi], ex);
}

// ------------- weighted scatter hyperedges -> vertices -------------
__global__ void __launch_bounds__(256) k_scatter_vertex(const float* __restrict__ Xe,
    const int* __restrict__ vertex, const int* __restrict__ edges,
    const float* __restrict__ aexp, const float* __restrict__ asum,
    float* __restrict__ Xv) {
  const int e = blockIdx.x;
  const int t = threadIdx.x;
  const int h = t >> 5;
  const int v = vertex[e];
  const int m = edges[e];
  const float attn = aexp[(size_t)e * HH + h] / (asum[(size_t)v * HH + h] + 1e-16f);
  atomicAdd(&Xv[(size_t)v * HC + t], Xe[(size_t)m * HC + t] * attn);
}

// ------------- exact GELU -------------
__global__ void k_gelu(const float* __restrict__ Xv, float* __restrict__ out) {
  size_t i = (size_t)blockIdx.x * blockDim.x + threadIdx.x;
  if (i >= (size_t)NV * HC) return;
  float x = Xv[i];
  out[i] = 0.5f * x * (1.0f + erff(x * 0.70710678118654752f));
}

extern "C" void kernel_launch(void* const* d_in, const int* in_sizes, int n_in,
                              void* d_out, int out_size, void* d_ws, size_t ws_size,
                              hipStream_t stream) {
  const float* X      = (const float*)d_in[0];
  const float* W      = (const float*)d_in[1];
  const float* att    = (const float*)d_in[2];
  const int*   vertex = (const int*)d_in[3];
  const int*   edges  = (const int*)d_in[4];
  float* ws  = (float*)d_ws;
  float* out = (float*)d_out;

  float* X0      = ws + OFF_X0;
  float* Xe      = ws + OFF_XE;
  float* Xv      = ws + OFF_XV;
  float* cnt     = ws + OFF_CNT;
  float* asum    = ws + OFF_ASUM;
  float* amax    = ws + OFF_AMAX;
  float* alpha_e = ws + OFF_AE;
  float* alphaE  = ws + OFF_AEXP;

  k_init<<<2048, 256, 0, stream>>>(ws);
  k_gemm<<<NV / 16, 256, 0, stream>>>(X, W, X0);
  k_scatter_edge<<<EI, 256, 0, stream>>>(X0, vertex, edges, Xe, cnt);
  k_edge_alpha<<<ME, 256, 0, stream>>>(Xe, cnt, att, alpha_e);
  const int EH = EI * HH;
  k_leaky_amax<<<(EH + 255) / 256, 256, 0, stream>>>(alpha_e, vertex, edges, alphaE, amax);
  k_exp_asum<<<(EH + 255) / 256, 256, 0, stream>>>(alphaE, vertex, amax, asum);
  k_scatter_vertex<<<EI, 256, 0, stream>>>(Xe, vertex, edges, alphaE, asum, Xv);
  k_gelu<<<((NV * HC) + 255) / 256, 256, 0, stream>>>(Xv, out);
}